// BatchDynamicSoftLabelAssigner_56178172231976
// MI455X (gfx1250) — compile-verified
//
#include <hip/hip_runtime.h>
#include <hip/hip_bf16.h>

// Problem constants (match reference)
constexpr int kB = 16;
constexpr int kN = 8400;
constexpr int kG = 100;
constexpr int kC = 80;      // NUM_CLASSES
constexpr float kEPS = 1e-7f;
constexpr float kINF = 1e8f;
constexpr float kLN10 = 2.302585092994046f;

typedef float v2f __attribute__((ext_vector_type(2)));
typedef float v8f __attribute__((ext_vector_type(8)));

// Builtin expects (addrspace(1) int*, addrspace(3) int*, imm offset, imm cpol)
#define GPTR(p) ((__attribute__((address_space(1))) int*)(p))
#define LPTR(p) ((__attribute__((address_space(3))) int*)(p))

// ---------------------------------------------------------------------------
// Kernel 1: pairwise cost + |giou| matrix. One wave (32 lanes) per 16x16 (n,g)
// tile. The soft-center distance cross-term pt.ctr is a (16x4)@(4x16) fp32
// WMMA (K padded with zeros). D layout: vgpr v, lane l -> n_local = v + 8*(l>=16),
// g_local = l&15 -- each lane then finishes the elementwise GIoU/BCE math for
// its 8 pairs.
// ---------------------------------------------------------------------------
__global__ __launch_bounds__(32) void pair_cost_kernel(
    const float* __restrict__ pred_bboxes,  // (B,N,4)
    const float* __restrict__ pred_scores,  // (B,N,C)
    const float* __restrict__ priors,       // (N,4): x,y,stride,stride
    const int*   __restrict__ gt_labels,    // (B,G)
    const float* __restrict__ gt_bboxes,    // (B,G,4)
    const float* __restrict__ pad_flag,     // (B,G)
    float* __restrict__ cost,               // (B,N,G)
    float* __restrict__ iousM)              // (B,N,G)
{
  const int lane = threadIdx.x;           // 0..31
  const int n0 = blockIdx.x * 16;
  const int g0 = blockIdx.y * 16;
  const int b  = blockIdx.z;

  const int gl   = lane & 15;
  const int half = lane >> 4;
  const int g    = g0 + gl;
  const bool gvalid = (g < kG);

  // Per-lane GT data (lanes 16..31 mirror lanes 0..15)
  float gx0 = 0.f, gy0 = 0.f, gx1 = 0.f, gy1 = 0.f, pflag = 0.f;
  int label = 0;
  if (gvalid) {
    const float* gb = gt_bboxes + ((size_t)b * kG + g) * 4;
    gx0 = gb[0]; gy0 = gb[1]; gx1 = gb[2]; gy1 = gb[3];
    pflag = pad_flag[(size_t)b * kG + g];
    label = gt_labels[(size_t)b * kG + g];
  }
  const float gcx = 0.5f * (gx0 + gx1);
  const float gcy = 0.5f * (gy0 + gy1);
  const float gnorm = gcx * gcx + gcy * gcy;

  // A matrix (16x4 f32): lanes 0-15 hold K=0,1 (=x,y) in vgpr0,1; lanes 16-31
  // hold K=2,3 which are the zero padding.
  v2f A = {0.f, 0.f};
  if (half == 0) {
    const float* pr = priors + (size_t)(n0 + lane) * 4;
    A.x = pr[0];
    A.y = pr[1];
  }
  // B matrix (4x16 f32): rows K=0,1 = gt center x,y; rows K=2,3 zero.
  v2f Bm = {0.f, 0.f};
  if (half == 0 && gvalid) {
    Bm.x = gcx;
    Bm.y = gcy;
  }

  v8f acc = {};
  // D[n_local][g_local] = px*gcx + py*gcy  (exact fp32 dot product)
  acc = __builtin_amdgcn_wmma_f32_16x16x4_f32(
      /*neg_a=*/false, A, /*neg_b=*/false, Bm,
      /*c_mod=*/(short)0, acc, /*reuse_a=*/false, /*reuse_b=*/false);

  const float a2 = (gx1 - gx0) * (gy1 - gy0);  // gt area (loop invariant)

#pragma unroll
  for (int v = 0; v < 8; ++v) {
    const int n = n0 + v + 8 * half;
    const float* pr = priors + (size_t)n * 4;
    const float px = pr[0], py = pr[1], st = pr[2];
    const float* pb = pred_bboxes + ((size_t)b * kN + n) * 4;
    const float bx0 = pb[0], by0 = pb[1], bx1 = pb[2], by1 = pb[3];
    float logit = 0.f;
    if (gvalid) logit = pred_scores[((size_t)b * kN + n) * kC + label];

    // is_in: prior center strictly inside gt box, times pad flag
    const float d0 = px - gx0, d1 = py - gy0, d2_ = gx1 - px, d3 = gy1 - py;
    const float mind = fminf(fminf(d0, d1), fminf(d2_, d3));
    const float is_in = (mind > 0.f) ? pflag : 0.f;

    // soft center prior via WMMA dot product
    const float dot = acc[v];
    const float dd = fmaxf(px * px + py * py + gnorm - 2.f * dot, 0.f);
    const float dist = (sqrtf(dd) / st) * is_in;
    const float soft = __expf((dist - 3.0f) * kLN10) * is_in;  // 10^(dist-3)

    // GIoU(pred, gt)
    const float ltx = fmaxf(bx0, gx0), lty = fmaxf(by0, gy0);
    const float rbx = fminf(bx1, gx1), rby = fminf(by1, gy1);
    const float ow = fmaxf(rbx - ltx, 0.f), oh = fmaxf(rby - lty, 0.f);
    const float ov = ow * oh;
    const float a1 = (bx1 - bx0) * (by1 - by0);
    const float uni = fmaxf(a1 + a2 - ov, kEPS);
    const float iou = ov / uni;
    const float ex0 = fminf(bx0, gx0), ey0 = fminf(by0, gy0);
    const float ex1 = fmaxf(bx1, gx1), ey1 = fmaxf(by1, gy1);
    const float ew = fmaxf(ex1 - ex0, 0.f), eh = fmaxf(ey1 - ey0, 0.f);
    const float enc = fmaxf(ew * eh, kEPS);
    const float giou = iou - (enc - uni) / enc;

    const float piou = fabsf(giou * is_in);
    const float iou_cost = -__logf(piou + kEPS) * 3.0f;

    // soft BCE classification cost
    const float sig = 1.f / (1.f + __expf(-logit));
    const float scale = piou - sig;
    const float asc = fabsf(scale);
    const float bce = fmaxf(logit, 0.f) - logit * piou + log1pf(__expf(-fabsf(logit)));
    const float cls = bce * __expf(asc - 1.f) * asc;

    float cst = cls + iou_cost + soft;
    cst = (is_in > 0.f) ? cst : kINF;

    if (gvalid) {
      const size_t idx = ((size_t)b * kN + n) * kG + g;
      cost[idx] = cst;
      iousM[idx] = piou;
    }
  }
}

// ---------------------------------------------------------------------------
// Async global->LDS staging of one strided column (stride G floats).
// Uses GLOBAL_LOAD_ASYNC_TO_LDS_B32 (ASYNCcnt) when the toolchain exposes the
// builtin; falls back to plain VGPR staging otherwise.
// ---------------------------------------------------------------------------
__device__ __forceinline__ void stage_column_async(
    float* lds_col, const float* gcol, int t) {
#if __has_builtin(__builtin_amdgcn_global_load_async_to_lds_b32)
  for (int n = t; n < kN; n += 256) {
    __builtin_amdgcn_global_load_async_to_lds_b32(
        GPTR(gcol + (size_t)n * kG), LPTR(lds_col + n), 0, 0);
  }
#if __has_builtin(__builtin_amdgcn_s_wait_asynccnt)
  __builtin_amdgcn_s_wait_asynccnt(0);
#else
  asm volatile("s_wait_asynccnt 0" ::: "memory");
#endif
#else
  for (int n = t; n < kN; n += 256) lds_col[n] = gcol[(size_t)n * kG];
#endif
}

// ---------------------------------------------------------------------------
// Kernel 2: per (b,g) column — dyn_k from top-13 ious, then mark the dyn_k
// smallest-cost anchors (stable index tie-break == argsort ranks < dyn_k).
// Column staged in LDS (33.6 KB) via async-to-LDS. Also emits area_dist.
// ---------------------------------------------------------------------------
__global__ __launch_bounds__(256) void select_kernel(
    const float* __restrict__ cost,
    const float* __restrict__ iousM,
    const float* __restrict__ gt_area,    // (B,G)
    const float* __restrict__ area_rule,  // (4,)
    const float* __restrict__ pad_flag,   // (B,G)
    unsigned char* __restrict__ matchM,   // (B,N,G) bytes
    float* __restrict__ area_out)         // (3,B,G)
{
  __shared__ float col[kN];
  __shared__ float rv[256];
  __shared__ int   ri[256];
  __shared__ float s_sum;
  __shared__ int   s_k;

  const int g = blockIdx.x;
  const int b = blockIdx.y;
  const int t = threadIdx.x;

  // ---- phase 1: sum of 13 largest ious -> dyn_k ----
  stage_column_async(col, iousM + (size_t)b * kN * kG + g, t);
  if (t == 0) s_sum = 0.f;
  __syncthreads();

  for (int it = 0; it < 13; ++it) {
    float best = -1.f;
    int bi = kN;
    for (int n = t; n < kN; n += 256) {
      const float vv = col[n];
      if (vv > best) { best = vv; bi = n; }
    }
    rv[t] = best; ri[t] = bi;
    __syncthreads();
    for (int s = 128; s; s >>= 1) {
      if (t < s) {
        if (rv[t + s] > rv[t]) { rv[t] = rv[t + s]; ri[t] = ri[t + s]; }
      }
      __syncthreads();
    }
    if (t == 0) {
      s_sum += fmaxf(rv[0], 0.f);
      col[ri[0]] = -1.f;  // exclude (ious are >= 0)
    }
    __syncthreads();
  }
  if (t == 0) {
    int k = (int)s_sum;  // trunc, matches astype(int32)
    if (k < 1) k = 1;
    if (k > 13) k = 13;  // sum of 13 ious in [0,1] can't exceed 13
    s_k = k;
  }
  __syncthreads();
  const int k = s_k;

  // ---- phase 2: mark dyn_k smallest costs (stable tie-break by index) ----
  stage_column_async(col, cost + (size_t)b * kN * kG + g, t);
  __syncthreads();

  for (int it = 0; it < k; ++it) {
    float best = 3.4e38f;
    int bi = kN;
    for (int n = t; n < kN; n += 256) {
      const float vv = col[n];
      if (vv < best) { best = vv; bi = n; }  // ascending scan keeps lowest n on tie
    }
    rv[t] = best; ri[t] = bi;
    __syncthreads();
    for (int s = 128; s; s >>= 1) {
      if (t < s) {
        if (rv[t + s] < rv[t] || (rv[t + s] == rv[t] && ri[t + s] < ri[t])) {
          rv[t] = rv[t + s]; ri[t] = ri[t + s];
        }
      }
      __syncthreads();
    }
    if (t == 0) {
      const int n = ri[0];
      matchM[((size_t)b * kN + n) * kG + g] = 1;
      col[n] = 3.4e38f;  // exclude
    }
    __syncthreads();
  }

  // ---- area_dist (3,B,G) ----
  if (t == 0) {
    const float ga = gt_area[(size_t)b * kG + g];
    const bool valid = pad_flag[(size_t)b * kG + g] > 0.f;
    float rr[4];
#pragma unroll
    for (int i = 0; i < 4; ++i) { const float r = area_rule[i]; rr[i] = r * r; }
#pragma unroll
    for (int i = 0; i < 3; ++i) {
      area_out[((size_t)i * kB + b) * kG + g] =
          (ga >= rr[i] && ga < rr[i + 1] && valid) ? 1.f : 0.f;
    }
  }
}

// ---------------------------------------------------------------------------
// Kernel 3: per (b,n) anchor — resolve multi-match via argmin cost, emit
// labels/bboxes/metrics/fg and accumulate per-class counts.
// ---------------------------------------------------------------------------
__global__ __launch_bounds__(256) void match_kernel(
    const float* __restrict__ cost,
    const float* __restrict__ iousM,
    const unsigned char* __restrict__ matchM,
    const int* __restrict__ gt_labels,    // (B,G)
    const float* __restrict__ gt_bboxes,  // (B,G,4)
    float* __restrict__ out_labels,       // (B,N)
    float* __restrict__ out_bboxes,       // (B,N,4)
    float* __restrict__ out_metrics,      // (B,N)
    float* __restrict__ out_fg,           // (B,N)
    int*   __restrict__ ws_label,
    float* __restrict__ ws_metric,
    float* __restrict__ ws_fg,
    float* __restrict__ labels_num_acc)   // (80,) float atomics
{
  const size_t i = (size_t)blockIdx.x * 256 + threadIdx.x;  // b*N + n
  if (i >= (size_t)kB * kN) return;
  const int b = (int)(i / kN);

  const unsigned char* row = matchM + i * kG;
  __builtin_prefetch(row, 0, 0);  // global_prefetch_b8
  int cnt = 0, mg = -1;
  for (int g = 0; g < kG; ++g) {
    if (row[g]) { ++cnt; if (mg < 0) mg = g; }
  }
  if (cnt > 1) {  // multi-match -> argmin over cost row (first on ties)
    const float* crow = cost + i * kG;
    float best = crow[0];
    mg = 0;
    for (int g = 1; g < kG; ++g) {
      const float vv = crow[g];
      if (vv < best) { best = vv; mg = g; }
    }
  }
  const bool fg = (cnt > 0);
  float metric = 0.f;
  int lab = kC;
  float bx0 = 0.f, by0 = 0.f, bx1 = 0.f, by1 = 0.f;
  if (fg) {
    metric = iousM[i * kG + mg];
    lab = gt_labels[(size_t)b * kG + mg];
    const float* gb = gt_bboxes + ((size_t)b * kG + mg) * 4;
    bx0 = gb[0]; by0 = gb[1]; bx1 = gb[2]; by1 = gb[3];
    atomicAdd(&labels_num_acc[lab], 1.0f);
  }
  out_labels[i] = (float)lab;
  out_metrics[i] = fg ? metric : 0.f;
  out_fg[i] = fg ? 1.f : 0.f;
  float* ob = out_bboxes + i * 4;
  ob[0] = bx0; ob[1] = by0; ob[2] = bx1; ob[3] = by1;
  ws_label[i] = lab;
  ws_metric[i] = metric;
  ws_fg[i] = fg ? 1.f : 0.f;
}

// ---------------------------------------------------------------------------
// Kernel 4: class histogram -> dynamic class weights (single tiny block).
// ---------------------------------------------------------------------------
__global__ __launch_bounds__(128) void weight_kernel(
    const float* __restrict__ acc,  // (80,)
    float* __restrict__ out_num,    // (80,)
    float* __restrict__ out_cw,     // (80,)
    float* __restrict__ ws_cw)      // (80,)
{
  __shared__ float s[kC];
  __shared__ float maxc;
  const int t = threadIdx.x;
  if (t < kC) s[t] = acc[t];
  __syncthreads();
  if (t == 0) {
    float m = s[0];
    for (int c = 1; c < kC; ++c) m = fmaxf(m, s[c]);
    maxc = m;
  }
  __syncthreads();
  if (t < kC) {
    const float v = s[t];
    const float w = (v > 0.f) ? (maxc / fmaxf(v, 1.f)) : 1.f;
    out_num[t] = v;
    out_cw[t] = w;
    ws_cw[t] = w;
  }
}

// ---------------------------------------------------------------------------
// Kernel 5: broadcast (B,N,C) assigned_scores and label weights.
// ---------------------------------------------------------------------------
__global__ __launch_bounds__(256) void scores_weights_kernel(
    const int*   __restrict__ ws_label,
    const float* __restrict__ ws_metric,
    const float* __restrict__ ws_fg,
    const float* __restrict__ ws_cw,
    float* __restrict__ out_weights,  // (B,N,C)
    float* __restrict__ out_scores)   // (B,N,C)
{
  const size_t idx = (size_t)blockIdx.x * 256 + threadIdx.x;
  if (idx >= (size_t)kB * kN * kC) return;
  const int c = (int)(idx % kC);
  const size_t i = idx / kC;
  const bool fg = ws_fg[i] > 0.f;
  out_weights[idx] = fg ? ws_cw[c] : 1.f;
  out_scores[idx] = (fg && ws_label[i] == c) ? ws_metric[i] : 0.f;
}

// ---------------------------------------------------------------------------
extern "C" void kernel_launch(void* const* d_in, const int* in_sizes, int n_in,
                              void* d_out, int out_size, void* d_ws, size_t ws_size,
                              hipStream_t stream) {
  const float* pred_bboxes = (const float*)d_in[0];
  const float* pred_scores = (const float*)d_in[1];
  const float* priors      = (const float*)d_in[2];
  const int*   gt_labels   = (const int*)d_in[3];
  const float* gt_bboxes   = (const float*)d_in[4];
  const float* pad_flag    = (const float*)d_in[5];
  const float* gt_area     = (const float*)d_in[6];
  const float* area_rule   = (const float*)d_in[7];

  float* out = (float*)d_out;
  const size_t NB = (size_t)kB * kN;  // 134400
  const size_t off_labels  = 0;
  const size_t off_weights = off_labels + NB;
  const size_t off_bboxes  = off_weights + NB * kC;
  const size_t off_scores  = off_bboxes + NB * 4;
  const size_t off_metrics = off_scores + NB * kC;
  const size_t off_fg      = off_metrics + NB;
  const size_t off_num     = off_fg + NB;
  const size_t off_cw      = off_num + kC;
  const size_t off_area    = off_cw + kC;

  // Workspace carve (~123 MB): cost, ious, match bytes, per-anchor scalars.
  float* cost = (float*)d_ws;
  float* iousM = cost + NB * kG;
  unsigned char* matchM = (unsigned char*)(iousM + NB * kG);
  const size_t matchBytes = NB * kG;  // 13.44 MB, 16B-aligned
  int*   ws_label  = (int*)(matchM + matchBytes);
  float* ws_metric = (float*)(ws_label + NB);
  float* ws_fg     = ws_metric + NB;
  float* acc       = ws_fg + NB;      // 80 floats
  float* ws_cw     = acc + kC;        // 80 floats

  (void)hipMemsetAsync(matchM, 0, matchBytes, stream);
  (void)hipMemsetAsync(acc, 0, kC * sizeof(float), stream);

  // 1) pairwise cost / ious (WMMA f32 16x16x4 for the center dot products)
  dim3 grid1(kN / 16, (kG + 15) / 16, kB);
  pair_cost_kernel<<<grid1, 32, 0, stream>>>(
      pred_bboxes, pred_scores, priors, gt_labels, gt_bboxes, pad_flag,
      cost, iousM);

  // 2) per-(b,g) dynamic-k selection + area_dist
  dim3 grid2(kG, kB);
  select_kernel<<<grid2, 256, 0, stream>>>(
      cost, iousM, gt_area, area_rule, pad_flag, matchM, out + off_area);

  // 3) per-anchor match resolution
  const int nAnchorBlocks = (int)((NB + 255) / 256);
  match_kernel<<<nAnchorBlocks, 256, 0, stream>>>(
      cost, iousM, matchM, gt_labels, gt_bboxes,
      out + off_labels, out + off_bboxes, out + off_metrics, out + off_fg,
      ws_label, ws_metric, ws_fg, acc);

  // 4) class weights
  weight_kernel<<<1, 128, 0, stream>>>(acc, out + off_num, out + off_cw, ws_cw);

  // 5) (B,N,C) scores + weights broadcast
  const size_t total = NB * kC;
  const int nBC = (int)((total + 255) / 256);
  scores_weights_kernel<<<nBC, 256, 0, stream>>>(
      ws_label, ws_metric, ws_fg, ws_cw, out + off_weights, out + off_scores);
}